// OpticalFlow_27273042330376
// MI455X (gfx1250) — compile-verified
//
#include <hip/hip_runtime.h>
#include <hip/hip_bf16.h>
#include <cstdint>

typedef float v2f __attribute__((ext_vector_type(2)));
typedef float v8f __attribute__((ext_vector_type(8)));

#define HW_  4096   // 64*64
#define CCH  256    // channels
#define HDIM 64

// ---------------------------------------------------------------------------
// Kernel 1: all-pairs correlation  corr[b][i][j] = (1/16) * sum_c f1[b][c][i]*f2[b][c][j]
// f1/f2 layout: (B, C, HW) row-major, channel stride = HW.
// One wave computes a 16(M) x 64(N) tile with V_WMMA_F32_16X16X4_F32;
// 8 waves/block -> 128 x 64 block tile.  Exact fp32 accumulation.
// The shared 4x64 B tiles are staged per-block into LDS via the CDNA5
// async global->LDS path (ASYNCcnt), removing the 8x redundant B reads.
// ---------------------------------------------------------------------------
__global__ __launch_bounds__(256)
void corr_gemm_kernel(const float* __restrict__ f1, const float* __restrict__ f2,
                      float* __restrict__ corr)
{
    constexpr int KCHUNK  = 32;
    constexpr int BSTRIDE = 72;   // padded row stride: 2*72 % 64 == 16 -> half-waves on disjoint banks
    __shared__ __align__(16) float Bs[KCHUNK * BSTRIDE];

    const int tid  = threadIdx.x;
    const int wave = tid >> 5;
    const int lane = tid & 31;
    const int b    = blockIdx.z;
    const int i0   = blockIdx.y * 128 + wave * 16;   // M block
    const int j0   = blockIdx.x * 64;                // N block

    // A fragment (16x4 f32): lanes 0-15 -> M=lane, VGPR0/1 = K+0/K+1;
    // lanes 16-31 -> M=lane-16, VGPR0/1 = K+2/K+3.  B symmetric over N.
    const int row  = lane & 15;
    const int koff = (lane >> 4) << 1;               // 0 or 2

    const float* __restrict__ Abase = f1 + (size_t)b * CCH * HW_ + i0 + row;
    const float* __restrict__ Bglob = f2 + (size_t)b * CCH * HW_ + j0;

    // Cooperative staging map: thread -> (k-row sr / sr+16, 4-float column sc)
    const int sr = tid >> 4;                 // 0..15
    const int sc = (tid & 15) << 2;          // 0..60 step 4
    // Low 32 bits of the flat address of an LDS location == wave-relative LDS byte offset.
    const uint32_t lds0 = (uint32_t)(uintptr_t)(void*)&Bs[sr * BSTRIDE + sc];
    const uint32_t lds1 = (uint32_t)(uintptr_t)(void*)&Bs[(sr + 16) * BSTRIDE + sc];

    v8f c0 = {}; v8f c1 = {}; v8f c2 = {}; v8f c3 = {};
    const int n = lane & 15;

    for (int kc = 0; kc < CCH; kc += KCHUNK) {
        const uint64_t g0 = (uint64_t)(uintptr_t)(Bglob + (size_t)(kc + sr) * HW_ + sc);
        const uint64_t g1 = (uint64_t)(uintptr_t)(Bglob + (size_t)(kc + sr + 16) * HW_ + sc);
        // Async DMA of the 32x64 B chunk into LDS; each wave waits its own
        // ASYNCcnt, so barrier arrival implies the whole chunk has landed.
        asm volatile(
            "global_load_async_to_lds_b128 %0, %2, off\n\t"
            "global_load_async_to_lds_b128 %1, %3, off\n\t"
            "s_wait_asynccnt 0"
            :
            : "v"(lds0), "v"(lds1), "v"(g0), "v"(g1)
            : "memory");
        __syncthreads();

        #pragma unroll
        for (int kk = 0; kk < KCHUNK; kk += 4) {
            const size_t ko = (size_t)(kc + kk + koff) * HW_;
            v2f a;  a.x = Abase[ko];  a.y = Abase[ko + HW_];
            const float* __restrict__ br0 = &Bs[(kk + koff) * BSTRIDE + n];
            const float* __restrict__ br1 = &Bs[(kk + koff + 1) * BSTRIDE + n];
            v2f b0; b0.x = br0[0];  b0.y = br1[0];
            v2f b1; b1.x = br0[16]; b1.y = br1[16];
            v2f b2; b2.x = br0[32]; b2.y = br1[32];
            v2f b3; b3.x = br0[48]; b3.y = br1[48];
            c0 = __builtin_amdgcn_wmma_f32_16x16x4_f32(false, a, false, b0, (short)0, c0, false, false);
            c1 = __builtin_amdgcn_wmma_f32_16x16x4_f32(false, a, false, b1, (short)0, c1, false, false);
            c2 = __builtin_amdgcn_wmma_f32_16x16x4_f32(false, a, false, b2, (short)0, c2, false, false);
            c3 = __builtin_amdgcn_wmma_f32_16x16x4_f32(false, a, false, b3, (short)0, c3, false, false);
        }
        __syncthreads();
    }

    // C/D layout: lane L holds N = L&15, rows M = r + 8*(L>=16), r = VGPR 0..7
    const float scale = 0.0625f;                     // 1/sqrt(256)
    const int   mhi   = (lane >> 4) << 3;
    const size_t outb = (((size_t)b * HW_) + i0 + mhi) * HW_ + j0 + n;
    #pragma unroll
    for (int r = 0; r < 8; ++r) {
        const size_t o = outb + (size_t)r * HW_;
        corr[o]      = c0[r] * scale;
        corr[o + 16] = c1[r] * scale;
        corr[o + 32] = c2[r] * scale;
        corr[o + 48] = c3[r] * scale;
    }
}

// ---------------------------------------------------------------------------
// Kernel 2: 2x2 average pool on (N, Hin, Hin) -> (N, Hin/2, Hin/2)
// ---------------------------------------------------------------------------
__global__ void avgpool2_kernel(const float* __restrict__ in, float* __restrict__ out,
                                int n_total, int Hin)
{
    const int idx = blockIdx.x * blockDim.x + threadIdx.x;
    if (idx >= n_total) return;
    const int Hout = Hin >> 1;
    const int hw   = Hout * Hout;
    const int n    = idx / hw;
    const int rem  = idx - n * hw;
    const int y    = rem / Hout;
    const int x    = rem - y * Hout;
    const float* p = in + (((size_t)n * Hin) + 2 * y) * Hin + 2 * x;
    out[idx] = 0.25f * (p[0] + p[1] + p[Hin] + p[Hin + 1]);
}

// ---------------------------------------------------------------------------
// Kernel 3: bilinear lookup, grid_sample(zeros, align_corners=True) semantics.
// One thread per output element, output layout (B, 4*81, H, W).
// ---------------------------------------------------------------------------
__device__ __forceinline__ float bilin_zeros(const float* __restrict__ img, int S,
                                             float xs, float ys)
{
    const float x0f = floorf(xs), y0f = floorf(ys);
    const float wx1 = xs - x0f, wx0 = 1.0f - wx1;
    const float wy1 = ys - y0f, wy0 = 1.0f - wy1;
    const int xi = (int)x0f, yi = (int)y0f;

    float v00 = 0.f, v10 = 0.f, v01 = 0.f, v11 = 0.f;
    const bool xin0 = (xi     >= 0) & (xi     <= S - 1);
    const bool xin1 = (xi + 1 >= 0) & (xi + 1 <= S - 1);
    const bool yin0 = (yi     >= 0) & (yi     <= S - 1);
    const bool yin1 = (yi + 1 >= 0) & (yi + 1 <= S - 1);
    if (yin0) {
        const float* r = img + (size_t)yi * S;
        if (xin0) v00 = r[xi];
        if (xin1) v10 = r[xi + 1];
    }
    if (yin1) {
        const float* r = img + (size_t)(yi + 1) * S;
        if (xin0) v01 = r[xi];
        if (xin1) v11 = r[xi + 1];
    }
    return wy0 * (wx0 * v00 + wx1 * v10) + wy1 * (wx0 * v01 + wx1 * v11);
}

__global__ void lookup_kernel(const float* __restrict__ p0, const float* __restrict__ p1,
                              const float* __restrict__ p2, const float* __restrict__ p3,
                              const float* __restrict__ coords, float* __restrict__ out,
                              int total)
{
    const int tid = blockIdx.x * blockDim.x + threadIdx.x;
    if (tid >= total) return;

    const int w = tid & 63;
    const int h = (tid >> 6) & 63;
    const int c = (tid >> 12) % 324;
    const int b = tid / (324 * HW_);
    const int l = c / 81;
    const int k = c - l * 81;
    const int n = b * HW_ + h * HDIM + w;

    const size_t cbase = (size_t)b * 2 * HW_ + (size_t)h * HDIM + w;
    const float x = coords[cbase];           // channel 0 = x
    const float y = coords[cbase + HW_];     // channel 1 = y

    const float inv = 1.0f / (float)(1 << l);
    const int   S   = HDIM >> l;
    const float* img = (l == 0 ? p0 : l == 1 ? p1 : l == 2 ? p2 : p3) + (size_t)n * S * S;

    // Reference quirk: dy grid (slow index) added to x, dx grid (fast) to y.
    const float xs = x * inv + (float)(k / 9 - 4);
    const float ys = y * inv + (float)(k % 9 - 4);

    out[tid] = bilin_zeros(img, S, xs, ys);
}

// ---------------------------------------------------------------------------
extern "C" void kernel_launch(void* const* d_in, const int* in_sizes, int n_in,
                              void* d_out, int out_size, void* d_ws, size_t ws_size,
                              hipStream_t stream)
{
    (void)in_sizes; (void)n_in; (void)ws_size;

    const float* f1     = (const float*)d_in[0];
    const float* f2     = (const float*)d_in[1];
    const float* coords = (const float*)d_in[2];
    float* out = (float*)d_out;

    // Workspace layout (floats): corr 33.5M | L1 8.4M | L2 2.1M | L3 0.52M  (~178 MB)
    float* corr = (float*)d_ws;
    float* l1   = corr + (size_t)2 * HW_ * HW_;          // 33,554,432
    float* l2   = l1   + (size_t)2 * HW_ * 32 * 32;      //  8,388,608
    float* l3   = l2   + (size_t)2 * HW_ * 16 * 16;      //  2,097,152

    // GEMM: grid = (N tiles of 64, M tiles of 128, B)
    dim3 gg(HW_ / 64, HW_ / 128, 2);
    corr_gemm_kernel<<<gg, 256, 0, stream>>>(f1, f2, corr);

    const int n1 = 2 * HW_ * 32 * 32;
    avgpool2_kernel<<<(n1 + 255) / 256, 256, 0, stream>>>(corr, l1, n1, 64);
    const int n2 = 2 * HW_ * 16 * 16;
    avgpool2_kernel<<<(n2 + 255) / 256, 256, 0, stream>>>(l1, l2, n2, 32);
    const int n3 = 2 * HW_ * 8 * 8;
    avgpool2_kernel<<<(n3 + 255) / 256, 256, 0, stream>>>(l2, l3, n3, 16);

    const int total = out_size;              // 2 * 324 * 64 * 64
    lookup_kernel<<<(total + 255) / 256, 256, 0, stream>>>(corr, l1, l2, l3, coords, out, total);
}